// DecoderRNN_38826504356488
// MI455X (gfx1250) — compile-verified
//
#include <hip/hip_runtime.h>
#include <math.h>

#define H 512
#define B 128
#define S 64
#define V 32000

typedef float v2f __attribute__((ext_vector_type(2)));
typedef float v8f __attribute__((ext_vector_type(8)));

// ---------------------------------------------------------------------------
// Fast activations on the native transcendental pipe.
// gfx1250 has V_TANH_F32; use the builtin when clang declares it, otherwise
// a single v_exp_f32 based form (exact saturation at +-inf).
// ---------------------------------------------------------------------------
static __device__ __forceinline__ float fast_tanh(float x) {
#if __has_builtin(__builtin_amdgcn_tanhf)
    return __builtin_amdgcn_tanhf(x);
#elif __has_builtin(__builtin_amdgcn_tanh_f32)
    return __builtin_amdgcn_tanh_f32(x);
#else
    float e = __expf(2.0f * x);              // v_exp_f32
    return 1.0f - 2.0f / (e + 1.0f);
#endif
}

static __device__ __forceinline__ float fast_sigmoid(float x) {
    return 1.0f / (1.0f + __expf(-x));       // v_exp_f32 + v_rcp path
}

// ---------------------------------------------------------------------------
// NT adjacent 16x16 output tiles of D = X(MxK, row-major) @ W(NxK, row-major)^T
// for one m-tile, accumulated over K with V_WMMA_F32_16X16X4_F32.
// A is loaded ONCE per k-step and reused for NT WMMAs (8x less A traffic).
// B tiles are addressed off one base pointer with compile-time immediate
// offsets (j*16*ldw*4 bytes fits the 24-bit IOFFSET field).
// A 16x4 layout: lanes 0-15 -> (M=lane, K=k0..k0+1), lanes 16-31 -> k0+2..k0+3
// B 4x16 mirrors A with N in place of M. C/D: acc[r] = D[m0 + 8*half + r][n0+l16].
// ---------------------------------------------------------------------------
template <int NT>
static __device__ __forceinline__ void wmma_xwt_multi(const float* __restrict__ X, int ldx,
                                                      const float* __restrict__ W, int ldw,
                                                      int m0, int n0, int K,
                                                      v8f (&acc)[NT]) {
    const int lane = threadIdx.x & 31;
    const int half = lane >> 4;
    const int l16  = lane & 15;
    const float* xr = X + (size_t)(m0 + l16) * ldx + 2 * half;
    const float* wr = W + (size_t)(n0 + l16) * ldw + 2 * half;
#pragma unroll 2
    for (int k = 0; k < K; k += 4) {
        v2f a = *(const v2f*)(xr + k);
#pragma unroll
        for (int j = 0; j < NT; ++j) {
            v2f b = *(const v2f*)(wr + k + (size_t)j * 16 * ldw);
            acc[j] = __builtin_amdgcn_wmma_f32_16x16x4_f32(false, a, false, b,
                                                           (short)0, acc[j], false, false);
        }
    }
}

// ---------------------------------------------------------------------------
// Embedding gather: embedded[b,h] = emb[ids[b], h]
// ---------------------------------------------------------------------------
__global__ void k_embed(const int* __restrict__ ids, const float* __restrict__ emb,
                        float* __restrict__ out) {
    int i = blockIdx.x * blockDim.x + threadIdx.x;   // 0 .. B*H-1
    int b = i >> 9;
    int h = i & (H - 1);
    out[i] = emb[(size_t)ids[b] * H + h];
}

// ---------------------------------------------------------------------------
// Generic D = X @ W^T (no bias), NT n-tiles per wave, exact tile grid.
// ---------------------------------------------------------------------------
template <int NT>
__global__ void __launch_bounds__(128)
k_gemm_xwt(const float* __restrict__ X, int ldx,
           const float* __restrict__ W, int ldw,
           float* __restrict__ D, int ldd, int Ng, int K) {
    int wave = (blockIdx.x * blockDim.x + threadIdx.x) >> 5;
    int mt = wave / Ng, g = wave % Ng;
    int m0 = mt * 16, n0 = g * 16 * NT;
    v8f acc[NT] = {};
    wmma_xwt_multi<NT>(X, ldx, W, ldw, m0, n0, K, acc);
    const int lane = threadIdx.x & 31, half = lane >> 4, l16 = lane & 15;
#pragma unroll
    for (int j = 0; j < NT; ++j) {
        int n = n0 + j * 16 + l16;
#pragma unroll
        for (int r = 0; r < 8; ++r)
            D[(size_t)(m0 + half * 8 + r) * ldd + n] = acc[j][r];
    }
}

// ---------------------------------------------------------------------------
// Fused attention scores: for 16 flattened (b,s) rows per wave,
//   scores[row] = sum_n v[n] * tanh( (enc @ W2^T)[row,n] + hW[b,n] + attn_b[n] )
// n is blocked in 4 groups of 128 columns (8 tiles); the [B*S,H] energy tensor
// is never materialized.
// ---------------------------------------------------------------------------
__global__ void __launch_bounds__(128)
k_scores(const float* __restrict__ enc, const float* __restrict__ attn_W,
         const float* __restrict__ hW, const float* __restrict__ attn_b,
         const float* __restrict__ vvec, float* __restrict__ scores) {
    __shared__ float red[4][32][8];
    const int wv   = threadIdx.x >> 5;                                   // wave in block
    const int wave = (blockIdx.x * blockDim.x + threadIdx.x) >> 5;       // global wave
    const int m0   = wave * 16;                                          // row base in [0, B*S)
    const int lane = threadIdx.x & 31, half = lane >> 4, l16 = lane & 15;

    float sacc[8] = {0.f, 0.f, 0.f, 0.f, 0.f, 0.f, 0.f, 0.f};
    for (int g = 0; g < H / 128; ++g) {                // 4 n-groups of 8 tiles
        const int n0 = g * 128;
        v8f acc[8] = {};
        // enc part uses attn_W columns [H, 2H)
        wmma_xwt_multi<8>(enc, H, attn_W + H, 2 * H, m0, n0, H, acc);
#pragma unroll
        for (int j = 0; j < 8; ++j) {
            const int n  = n0 + j * 16 + l16;
            const float bn = attn_b[n];
            const float vn = vvec[n];
#pragma unroll
            for (int r = 0; r < 8; ++r) {
                int row = m0 + half * 8 + r;
                int b   = row >> 6;                    // S == 64
                sacc[r] += vn * fast_tanh(acc[j][r] + hW[b * H + n] + bn);
            }
        }
    }
    // reduce the 16 lanes holding each row
#pragma unroll
    for (int r = 0; r < 8; ++r) red[wv][lane][r] = sacc[r];
    __syncthreads();
    if (half == 0) {
        int base = (l16 < 8) ? 0 : 16;
        int r    = l16 & 7;
        float s  = 0.f;
#pragma unroll
        for (int l = 0; l < 16; ++l) s += red[wv][base + l][r];
        scores[m0 + l16] = s;
    }
}

// ---------------------------------------------------------------------------
// Per-batch softmax over S=64 and context = attn_weights @ enc.  One block / b.
// ---------------------------------------------------------------------------
__global__ void k_softmax_ctx(const float* __restrict__ scores, const float* __restrict__ enc,
                              float* __restrict__ ctx) {
    __shared__ float w[S];
    __shared__ float red[S];
    int b = blockIdx.x, t = threadIdx.x;
    float sc = scores[b * S + t];
    red[t] = sc; __syncthreads();
    for (int off = S / 2; off > 0; off >>= 1) {
        if (t < off) red[t] = fmaxf(red[t], red[t + off]);
        __syncthreads();
    }
    float m = red[0]; __syncthreads();
    float e = __expf(sc - m);
    red[t] = e; __syncthreads();
    for (int off = S / 2; off > 0; off >>= 1) {
        if (t < off) red[t] += red[t + off];
        __syncthreads();
    }
    w[t] = e / red[0];
    __syncthreads();
    const float* encb = enc + (size_t)b * S * H;
    for (int hh = t; hh < H; hh += S) {
        float acc = 0.f;
#pragma unroll 8
        for (int s = 0; s < S; ++s) acc += w[s] * encb[s * H + hh];
        ctx[b * H + hh] = acc;
    }
}

// ---------------------------------------------------------------------------
// LSTM pre-activations: gates = [embedded|ctx] @ W_ih^T + h @ W_hh^T + b_ih + b_hh
// Three WMMA accumulation passes into 8 accumulator tiles per wave.
// ---------------------------------------------------------------------------
__global__ void __launch_bounds__(128)
k_gates(const float* __restrict__ embedded, const float* __restrict__ ctx,
        const float* __restrict__ h, const float* __restrict__ W_ih,
        const float* __restrict__ W_hh, const float* __restrict__ b_ih,
        const float* __restrict__ b_hh, float* __restrict__ gates) {
    int wave = (blockIdx.x * blockDim.x + threadIdx.x) >> 5;
    const int Ng = (4 * H) / 128;                 // 16 n-groups along 4H
    int mt = wave / Ng, g = wave % Ng;
    int m0 = mt * 16, n0 = g * 128;
    v8f acc[8] = {};
    wmma_xwt_multi<8>(embedded, H, W_ih,     2 * H, m0, n0, H, acc);  // rnn_in[:, :H]
    wmma_xwt_multi<8>(ctx,      H, W_ih + H, 2 * H, m0, n0, H, acc);  // rnn_in[:, H:]
    wmma_xwt_multi<8>(h,        H, W_hh,     H,     m0, n0, H, acc);  // recurrent
    const int lane = threadIdx.x & 31, half = lane >> 4, l16 = lane & 15;
#pragma unroll
    for (int j = 0; j < 8; ++j) {
        int n = n0 + j * 16 + l16;
        float bias = b_ih[n] + b_hh[n];
#pragma unroll
        for (int r = 0; r < 8; ++r)
            gates[(size_t)(m0 + half * 8 + r) * (4 * H) + n] = acc[j][r] + bias;
    }
}

// ---------------------------------------------------------------------------
// LSTM elementwise (PyTorch gate order i,f,g,o) -> h_new, c_new in d_out.
// ---------------------------------------------------------------------------
__global__ void k_lstm(const float* __restrict__ gates, const float* __restrict__ c0,
                       float* __restrict__ h_new, float* __restrict__ c_new) {
    int i = blockIdx.x * blockDim.x + threadIdx.x;   // 0 .. B*H-1
    int b = i >> 9, hh = i & (H - 1);
    const float* g = gates + (size_t)b * 4 * H;
    float gi = g[hh], gf = g[H + hh], gg = g[2 * H + hh], go = g[3 * H + hh];
    float cn = fast_sigmoid(gf) * c0[i] + fast_sigmoid(gi) * fast_tanh(gg);
    float hn = fast_sigmoid(go) * fast_tanh(cn);
    h_new[i] = hn;
    c_new[i] = cn;
}

// ---------------------------------------------------------------------------
// Vocab projection: prediction = h_new @ fc_W^T + fc_b.  Bandwidth-bound on
// the 65.5 MB fc_W stream (~2.8 us floor at 23.3 TB/s).  8 n-tiles per wave
// keep A (h_new) traffic at ~64 MB of L0/L2 hits instead of 512 MB.
// ---------------------------------------------------------------------------
__global__ void __launch_bounds__(128)
k_fc(const float* __restrict__ hn, const float* __restrict__ fc_W,
     const float* __restrict__ fc_b, float* __restrict__ pred) {
    int wave = (blockIdx.x * blockDim.x + threadIdx.x) >> 5;
    const int Ng = V / 128;                       // 250 n-groups of 8 tiles
    int mt = wave / Ng, g = wave % Ng;
    int m0 = mt * 16, n0 = g * 128;
    const int lane = threadIdx.x & 31, half = lane >> 4, l16 = lane & 15;
    __builtin_prefetch(fc_W + (size_t)(n0 + l16) * H, 0, 1);   // global_prefetch_b8
    v8f acc[8] = {};
    wmma_xwt_multi<8>(hn, H, fc_W, H, m0, n0, H, acc);
#pragma unroll
    for (int j = 0; j < 8; ++j) {
        int n = n0 + j * 16 + l16;
        float bias = fc_b[n];
#pragma unroll
        for (int r = 0; r < 8; ++r)
            pred[(size_t)(m0 + half * 8 + r) * V + n] = acc[j][r] + bias;
    }
}

// ---------------------------------------------------------------------------
extern "C" void kernel_launch(void* const* d_in, const int* in_sizes, int n_in,
                              void* d_out, int out_size, void* d_ws, size_t ws_size,
                              hipStream_t stream) {
    const int*   ids    = (const int*)d_in[0];
    const float* h0     = (const float*)d_in[1];    // [1,B,H] == [B,H]
    const float* c0     = (const float*)d_in[2];
    const float* enc    = (const float*)d_in[3];    // [B,S,H]
    const float* emb    = (const float*)d_in[4];
    const float* attn_W = (const float*)d_in[5];    // [H, 2H]
    const float* attn_b = (const float*)d_in[6];
    const float* vvec   = (const float*)d_in[7];
    const float* W_ih   = (const float*)d_in[8];    // [4H, 2H]
    const float* W_hh   = (const float*)d_in[9];    // [4H, H]
    const float* b_ih   = (const float*)d_in[10];
    const float* b_hh   = (const float*)d_in[11];
    const float* fc_W   = (const float*)d_in[12];   // [V, H]
    const float* fc_b   = (const float*)d_in[13];

    float* out   = (float*)d_out;
    float* pred  = out;                              // [B, V]
    float* h_new = out + (size_t)B * V;              // [1, B, H]
    float* c_new = h_new + (size_t)B * H;            // [1, B, H]

    // workspace layout (floats): ~1.87 MB total
    float* ws       = (float*)d_ws;
    float* embedded = ws;                  // B*H
    float* hW       = embedded + B * H;    // B*H   (h @ attn_W[:, :H]^T)
    float* scores   = hW + B * H;          // B*S
    float* ctx      = scores + B * S;      // B*H
    float* gates    = ctx + B * H;         // B*4H

    // 1. embedding gather
    k_embed<<<(B * H) / 256, 256, 0, stream>>>(ids, emb, embedded);
    // 2. hW = h @ attn_W[:, :H]^T  (Mt=8 x Ng=4 -> 32 waves)
    k_gemm_xwt<8><<<8, 128, 0, stream>>>(h0, H, attn_W, 2 * H, hW, H, 4, H);
    // 3. fused energy GEMM + tanh + dot(v) -> scores  (512 waves)
    k_scores<<<128, 128, 0, stream>>>(enc, attn_W, hW, attn_b, vvec, scores);
    // 4. softmax over S + context
    k_softmax_ctx<<<B, S, 0, stream>>>(scores, enc, ctx);
    // 5. LSTM gate GEMM  (Mt=8 x Ng=16 -> 128 waves)
    k_gates<<<32, 128, 0, stream>>>(embedded, ctx, h0, W_ih, W_hh, b_ih, b_hh, gates);
    // 6. LSTM elementwise -> h_new, c_new in d_out
    k_lstm<<<(B * H) / 256, 256, 0, stream>>>(gates, c0, h_new, c_new);
    // 7. vocab projection  (Mt=8 x Ng=250 -> 2000 waves)
    k_fc<<<500, 128, 0, stream>>>(h_new, fc_W, fc_b, pred);
}